// ResidualORN_58265526337597
// MI455X (gfx1250) — compile-verified
//
#include <hip/hip_runtime.h>

typedef __attribute__((ext_vector_type(2))) float v2f;
typedef __attribute__((ext_vector_type(8))) float v8f;

#define C_CH 256
#define NORI 8
#define NROT 2
#define BATCH 16
#define HDIM 64
#define WDIM 64
#define HWC  (HDIM * WDIM)        /* 4096 */
#define NTOT (BATCH * HWC)        /* 65536 */
#define KTAPS 9
#define WEFF_ELEMS (C_CH * KTAPS * C_CH)  /* 589824 */

// ---------------------------------------------------------------------------
// Kernel A: fold the NROT x NORI weight blocks into W_eff, stored t-major:
//   wt[(co*9 + t)*256 + ci] = sum_{r<2,o<8} weight_all[r*256+co][o*256+ci][t]
// ---------------------------------------------------------------------------
__global__ __launch_bounds__(256) void wreduce_kernel(
    const float* __restrict__ wall, float* __restrict__ wt) {
  int idx = blockIdx.x * 256 + threadIdx.x;
  if (idx >= WEFF_ELEMS) return;
  int ci = idx & 255;
  int t  = (idx >> 8) % KTAPS;
  int co = idx / (KTAPS * 256);
  const float* base = wall + (size_t)co * (NORI * C_CH * KTAPS)
                           + (size_t)ci * KTAPS + t;
  float s = 0.f;
#pragma unroll
  for (int r = 0; r < NROT; ++r)
#pragma unroll
    for (int o = 0; o < NORI; ++o)
      s += base[(size_t)r * C_CH * (NORI * C_CH * KTAPS)
                + (size_t)o * C_CH * KTAPS];
  wt[idx] = s;
}

__global__ void breduce_kernel(const float* __restrict__ bias,
                               float* __restrict__ beff) {
  int c = threadIdx.x;
  if (c < C_CH) beff[c] = (bias[c] + bias[C_CH + c]) * (float)NORI;
}

// ---------------------------------------------------------------------------
// Kernel B: implicit-GEMM 3x3 conv via V_WMMA_F32_16X16X4_F32.
//   M = C_out (256), N = B*H*W (65536), K = tap*256 + ci (2304).
// Block = 8 waves; block tile = 32 co x 256 n. Each wave register-blocks
// 2x2 WMMA subtiles (32 co x 32 n, four independent v8f accumulators).
// K loop unrolled x8 so global/LDS channel strides fold into immediate
// offsets (24-bit vmem / 16-bit ds offset fields) instead of VALU adds.
// Edge taps use clamped (always-legal) addresses + cndmask zeroing.
// Next tap's weight tile is prefetched (global_prefetch_b8) during compute.
// ---------------------------------------------------------------------------
#define COTILE 32
#define LDS_PITCH 260  /* stride % 64 banks == 4 -> conflict-free row reads */

__global__ __launch_bounds__(256) void conv_wmma_kernel(
    const float* __restrict__ x, const float* __restrict__ wt,
    const float* __restrict__ beff, const float* __restrict__ alphaPtr,
    float* __restrict__ out) {
  __shared__ float ldsA[COTILE * LDS_PITCH];

  const int coTile = blockIdx.x & 7;           // 8 tiles of 32 output chans
  const int nBlock = blockIdx.x >> 3;          // 256 blocks of 256 positions
  const int co0 = coTile * COTILE;
  const int wave = threadIdx.x >> 5;           // wave32
  const int lane = threadIdx.x & 31;
  const int khalf = (lane >> 4) << 1;          // lanes 16-31 hold K+2,K+3
  const int mrow = lane & 15;                  // A-matrix M row for this lane

  // Two N strips per wave (j = 0,1), 16 columns each.
  int bArr[2], hArr[2], wArr[2];
#pragma unroll
  for (int j = 0; j < 2; ++j) {
    const int n = nBlock * 256 + wave * 32 + j * 16 + (lane & 15);
    bArr[j] = n >> 12;
    hArr[j] = (n >> 6) & 63;
    wArr[j] = n & 63;
  }

  v8f acc00 = {}, acc01 = {}, acc10 = {}, acc11 = {};

  for (int t = 0; t < KTAPS; ++t) {
    __syncthreads();
    {
      const int col = threadIdx.x;             // 0..255 == ci, coalesced
#pragma unroll
      for (int j = 0; j < COTILE; ++j)
        ldsA[j * LDS_PITCH + col] =
            wt[((size_t)(co0 + j) * KTAPS + t) * 256 + col];
    }
    __syncthreads();

    // Prefetch next tap's 32x256 weight tile: 256 threads x 1 cacheline.
    if (t + 1 < KTAPS) {
      const int j = threadIdx.x >> 3;          // row 0..31
      const int seg = threadIdx.x & 7;         // 8 x 128B per 1KB row
      __builtin_prefetch(
          wt + ((size_t)(co0 + j) * KTAPS + (t + 1)) * 256 + seg * 32, 0, 1);
    }

    const int dh = t / 3 - 1, dw = t % 3 - 1;
    const float* xB[2];
    bool valid[2];
#pragma unroll
    for (int j = 0; j < 2; ++j) {
      const int hh = hArr[j] + dh, ww = wArr[j] + dw;
      valid[j] = (hh >= 0) & (hh < HDIM) & (ww >= 0) & (ww < WDIM);
      // Clamp to a legal in-image address; invalid lanes are zeroed below.
      const int hc = hh < 0 ? 0 : (hh > 63 ? 63 : hh);
      const int wc = ww < 0 ? 0 : (ww > 63 ? 63 : ww);
      xB[j] = x + (size_t)bArr[j] * (C_CH * HWC) + hc * WDIM + wc;
    }

#pragma unroll 8
    for (int ci0 = 0; ci0 < 256; ci0 += 4) {
      const int ck = ci0 + khalf;
      v2f a0, a1;
      a0.x = ldsA[mrow * LDS_PITCH + ck];
      a0.y = ldsA[mrow * LDS_PITCH + ck + 1];
      a1.x = ldsA[(mrow + 16) * LDS_PITCH + ck];
      a1.y = ldsA[(mrow + 16) * LDS_PITCH + ck + 1];
      v2f b0, b1;
      b0.x = valid[0] ? xB[0][(size_t)ck * HWC] : 0.f;
      b0.y = valid[0] ? xB[0][(size_t)(ck + 1) * HWC] : 0.f;
      b1.x = valid[1] ? xB[1][(size_t)ck * HWC] : 0.f;
      b1.y = valid[1] ? xB[1][(size_t)(ck + 1) * HWC] : 0.f;
      acc00 = __builtin_amdgcn_wmma_f32_16x16x4_f32(
          false, a0, false, b0, (short)0, acc00, false, false);
      acc01 = __builtin_amdgcn_wmma_f32_16x16x4_f32(
          false, a0, false, b1, (short)0, acc01, false, false);
      acc10 = __builtin_amdgcn_wmma_f32_16x16x4_f32(
          false, a1, false, b0, (short)0, acc10, false, false);
      acc11 = __builtin_amdgcn_wmma_f32_16x16x4_f32(
          false, a1, false, b1, (short)0, acc11, false, false);
    }
  }

  // Epilogue: out = x + (alpha/16) * (conv + b_eff)
  const float scale = alphaPtr[0] * (1.0f / (float)(NORI * NROT));
  const int moff = (lane >> 4) << 3;           // lanes 16-31 hold M = r+8
#pragma unroll
  for (int r = 0; r < 8; ++r) {
#pragma unroll
    for (int cj = 0; cj < 2; ++cj) {           // co subtile
      const int co = co0 + cj * 16 + r + moff;
      const float be = beff[co];
#pragma unroll
      for (int nj = 0; nj < 2; ++nj) {         // n subtile
        const size_t gi = ((size_t)bArr[nj] * C_CH + co) * HWC +
                          (size_t)(hArr[nj] * WDIM + wArr[nj]);
        const float a =
            (cj == 0) ? ((nj == 0) ? acc00[r] : acc01[r])
                      : ((nj == 0) ? acc10[r] : acc11[r]);
        out[gi] = x[gi] + scale * (a + be);
      }
    }
  }
}

// ---------------------------------------------------------------------------
// Kernel C: per-channel batch stats (training-mode BN) -> fused scale/shift.
// One block per channel; float4 loads + LDS tree reduction over 65536 elems.
// ---------------------------------------------------------------------------
__global__ __launch_bounds__(256) void stats_kernel(
    const float* __restrict__ out, const float* __restrict__ gamma,
    const float* __restrict__ beta, float* __restrict__ scaleArr,
    float* __restrict__ shiftArr) {
  __shared__ float s_sum[256];
  __shared__ float s_sq[256];
  const int c = blockIdx.x;
  float sum = 0.f, sq = 0.f;
  for (int i = threadIdx.x; i < NTOT / 4; i += 256) {
    const int bb = i >> 10;                    // i / 1024
    const int p4 = i & 1023;
    const float4 v =
        ((const float4*)out)[((size_t)bb * C_CH + c) * (HWC / 4) + p4];
    sum += v.x + v.y + v.z + v.w;
    sq += v.x * v.x + v.y * v.y + v.z * v.z + v.w * v.w;
  }
  s_sum[threadIdx.x] = sum;
  s_sq[threadIdx.x] = sq;
  __syncthreads();
  for (int s = 128; s > 0; s >>= 1) {
    if (threadIdx.x < s) {
      s_sum[threadIdx.x] += s_sum[threadIdx.x + s];
      s_sq[threadIdx.x] += s_sq[threadIdx.x + s];
    }
    __syncthreads();
  }
  if (threadIdx.x == 0) {
    const float mean = s_sum[0] * (1.0f / (float)NTOT);
    const float var = s_sq[0] * (1.0f / (float)NTOT) - mean * mean;
    const float inv = rsqrtf(var + 1e-5f);
    const float sc = gamma[c] * inv;
    scaleArr[c] = sc;
    shiftArr[c] = beta[c] - mean * sc;
  }
}

// ---------------------------------------------------------------------------
// Kernel D: y = out*scale[c] + shift[c], float4-vectorized, in place.
// ---------------------------------------------------------------------------
__global__ __launch_bounds__(256) void norm_kernel(
    float* __restrict__ out, const float* __restrict__ sc,
    const float* __restrict__ sh) {
  const size_t total4 = (size_t)NTOT * C_CH / 4;
  for (size_t i = (size_t)blockIdx.x * 256 + threadIdx.x; i < total4;
       i += (size_t)gridDim.x * 256) {
    const int c = (int)((i >> 10) & 255);  // (i*4 / 4096) % 256
    float4 v = ((float4*)out)[i];
    const float a = sc[c], b = sh[c];
    v.x = v.x * a + b;
    v.y = v.y * a + b;
    v.z = v.z * a + b;
    v.w = v.w * a + b;
    ((float4*)out)[i] = v;
  }
}

// ---------------------------------------------------------------------------
extern "C" void kernel_launch(void* const* d_in, const int* in_sizes, int n_in,
                              void* d_out, int out_size, void* d_ws,
                              size_t ws_size, hipStream_t stream) {
  const float* x     = (const float*)d_in[0];
  const float* wall  = (const float*)d_in[1];
  const float* bias  = (const float*)d_in[2];
  const float* alpha = (const float*)d_in[3];
  const float* gamma = (const float*)d_in[4];
  const float* beta  = (const float*)d_in[5];
  float* out = (float*)d_out;

  float* ws = (float*)d_ws;
  float* wt       = ws;                   // 589824 floats (2.25 MB)
  float* beff     = ws + WEFF_ELEMS;      // 256
  float* scaleArr = beff + 256;           // 256
  float* shiftArr = scaleArr + 256;       // 256

  wreduce_kernel<<<(WEFF_ELEMS + 255) / 256, 256, 0, stream>>>(wall, wt);
  breduce_kernel<<<1, 256, 0, stream>>>(bias, beff);
  conv_wmma_kernel<<<8 * (NTOT / 256), 256, 0, stream>>>(x, wt, beff, alpha,
                                                         out);
  stats_kernel<<<C_CH, 256, 0, stream>>>(out, gamma, beta, scaleArr, shiftArr);
  norm_kernel<<<2048, 256, 0, stream>>>(out, scaleArr, shiftArr);
}